// LutLinear_9921374454035
// MI455X (gfx1250) — compile-verified
//
#include <hip/hip_runtime.h>

// Problem constants (match reference).
#define KDIM 4096
#define NDIM 4096
#define WBIT 3
#define GROUP 128
#define NGROUPS 32            // KDIM / GROUP
#define NUM_APOT 3

// Kernel tiling.
#define THREADS 256
#define COLS_PER_THREAD 2
#define COLS_PER_BLOCK (THREADS * COLS_PER_THREAD)   // 512 columns per block
#define NBLK (NDIM / COLS_PER_BLOCK)                 // 8 column blocks
// grid = (NBLK, NGROUPS) -> 256 blocks (2048 wave32s), each: 512 columns x 1 group (128 k's)

#define NPOS4 (GROUP / 4)     // 32 positions for the 4-bit LUT
#define NPOS8 (GROUP / 8)     // 16 positions for the 8-bit LUT

// 2^a for a in [1,8): build the float exponent directly (no transcendental).
__device__ __forceinline__ float exp2i_small(int a) {
    return __int_as_float((a + 127) << 23);
}

// CDNA5 async global->LDS copy of one dword per lane (tracked on ASYNCcnt).
__device__ __forceinline__ void async_b32_to_lds(unsigned lds_byte_off, const float* gptr) {
    asm volatile("global_load_async_to_lds_b32 %0, %1, off"
                 :: "v"(lds_byte_off), "v"(gptr)
                 : "memory");
}
__device__ __forceinline__ void wait_asynccnt0() {
    asm volatile("s_wait_asynccnt 0" ::: "memory");
}

__global__ __launch_bounds__(THREADS)
void lutgemm_bcq_main(const float* __restrict__ x,
                      const int*   __restrict__ bw,     // [KDIM/32][WBIT][NDIM]
                      const int*   __restrict__ alpha,  // [NDIM][WBIT][NGROUPS][NUM_APOT]
                      float*       __restrict__ ws)     // [NGROUPS][NDIM]
{
    __shared__ float sX[GROUP];             // 0.5 KB: this block's x slice
    __shared__ float sLut4[NPOS4 * 16];     // 2 KB : 4-bit LUT (builder stage only)
    __shared__ float sLut8[NPOS8 * 256];    // 16 KB: 8-bit LUT (main loop)

    const int g  = blockIdx.y;        // group index (k-slice)
    const int nb = blockIdx.x;        // column block
    const int t  = threadIdx.x;

    // ---- Issue ALL global traffic up front so HBM latency hides behind LUT build ----

    // (1) x slice -> LDS via async-to-LDS (gfx1250 ASYNCcnt path).
    if (t < GROUP) {
        unsigned off = (unsigned)(size_t)(&sX[t]);   // low 32 bits of LDS aperture addr = LDS offset
        async_b32_to_lds(off, x + g * GROUP + t);
    }

    // (2) Packed weights: 3 bit-planes x 4 words x 2 columns (lane-consecutive -> coalesced b64).
    const int n0    = nb * COLS_PER_BLOCK + t * COLS_PER_THREAD;
    const int wbase = g * 4;   // 4 packed int32 words per bit-plane cover this group's 128 k's
    int2 q[WBIT][4];
    #pragma unroll
    for (int b = 0; b < WBIT; ++b) {
        #pragma unroll
        for (int w = 0; w < 4; ++w) {
            q[b][w] = *(const int2*)(bw + ((size_t)(wbase + w) * WBIT + b) * NDIM + n0);
        }
    }

    // (3) APoT exponents: per (col, b) a contiguous int3.
    int av[COLS_PER_THREAD][WBIT][NUM_APOT];
    #pragma unroll
    for (int c = 0; c < COLS_PER_THREAD; ++c) {
        #pragma unroll
        for (int b = 0; b < WBIT; ++b) {
            const int* ap = alpha + (((size_t)(n0 + c) * WBIT + b) * NGROUPS + g) * NUM_APOT;
            av[c][b][0] = ap[0];
            av[c][b][1] = ap[1];
            av[c][b][2] = ap[2];
        }
    }

    // ---- LUT construction (overlaps the in-flight global loads above) ----
    wait_asynccnt0();
    __syncthreads();

    // 4-bit LUT: lut4[p][m] = sum_{i<4} (m_i ? +x[4p+i] : -x[4p+i])
    for (int e = t; e < NPOS4 * 16; e += THREADS) {
        const int p = e >> 4;
        const int m = e & 15;
        const float x0 = sX[4 * p + 0];
        const float x1 = sX[4 * p + 1];
        const float x2 = sX[4 * p + 2];
        const float x3 = sX[4 * p + 3];
        sLut4[e] = ((m & 1) ? x0 : -x0) + ((m & 2) ? x1 : -x1) +
                   ((m & 4) ? x2 : -x2) + ((m & 8) ? x3 : -x3);
    }
    __syncthreads();

    // 8-bit LUT from pairs of 4-bit entries: lut8[p][m] = lut4[2p][m&15] + lut4[2p+1][m>>4]
    for (int e = t; e < NPOS8 * 256; e += THREADS) {
        const int p = e >> 8;
        const int m = e & 255;
        sLut8[e] = sLut4[(2 * p) * 16 + (m & 15)] + sLut4[(2 * p + 1) * 16 + (m >> 4)];
    }
    __syncthreads();

    // ---- Main loop: byte-indexed LUT lookups (1 ds_load per 8 weights) ----
    float acc0 = 0.0f, acc1 = 0.0f;

    #pragma unroll
    for (int b = 0; b < WBIT; ++b) {
        // 16 independent lookups per column, gathered into temps so the
        // scheduler can batch the ds_loads and collapse the dscnt waits.
        float a0[16], a1[16];
        #pragma unroll
        for (int w = 0; w < 4; ++w) {
            #pragma unroll
            for (int i = 0; i < 4; ++i) {
                const int base = (w * 4 + i) << 8;
                a0[w * 4 + i] = sLut8[base + ((q[b][w].x >> (8 * i)) & 255)];
                a1[w * 4 + i] = sLut8[base + ((q[b][w].y >> (8 * i)) & 255)];
            }
        }
        // Pairwise-tree reduction (breaks the serial fp-add chain).
        const float p0 = (((a0[0] + a0[1]) + (a0[2] + a0[3])) + ((a0[4] + a0[5]) + (a0[6] + a0[7]))) +
                         (((a0[8] + a0[9]) + (a0[10] + a0[11])) + ((a0[12] + a0[13]) + (a0[14] + a0[15])));
        const float p1 = (((a1[0] + a1[1]) + (a1[2] + a1[3])) + ((a1[4] + a1[5]) + (a1[6] + a1[7]))) +
                         (((a1[8] + a1[9]) + (a1[10] + a1[11])) + ((a1[12] + a1[13]) + (a1[14] + a1[15])));

        // APoT scale per (n, b, g): sum of NUM_APOT powers of two.
        const float s0 = exp2i_small(av[0][b][0]) + exp2i_small(av[0][b][1]) + exp2i_small(av[0][b][2]);
        const float s1 = exp2i_small(av[1][b][0]) + exp2i_small(av[1][b][1]) + exp2i_small(av[1][b][2]);
        acc0 = fmaf(s0, p0, acc0);
        acc1 = fmaf(s1, p1, acc1);
    }

    // Per-group contribution; reduced deterministically by the second kernel.
    *(float2*)(ws + (size_t)g * NDIM + n0) = make_float2(acc0, acc1);
}

__global__ __launch_bounds__(THREADS)
void lutgemm_bcq_reduce(const float* __restrict__ ws,
                        const float* __restrict__ bias,
                        float*       __restrict__ out)
{
    const int n = blockIdx.x * blockDim.x + threadIdx.x;
    if (n >= NDIM) return;
    float s = bias[n];
    #pragma unroll
    for (int g = 0; g < NGROUPS; ++g) {
        s += ws[(size_t)g * NDIM + n];
    }
    out[n] = s;
}

extern "C" void kernel_launch(void* const* d_in, const int* in_sizes, int n_in,
                              void* d_out, int out_size, void* d_ws, size_t ws_size,
                              hipStream_t stream) {
    const float* x     = (const float*)d_in[0];
    const int*   bw    = (const int*)  d_in[1];
    const int*   alpha = (const int*)  d_in[2];
    const float* bias  = (const float*)d_in[3];
    float* out = (float*)d_out;
    float* ws  = (float*)d_ws;   // needs NGROUPS*NDIM*4 = 512 KB

    dim3 grid(NBLK, NGROUPS);
    lutgemm_bcq_main<<<grid, THREADS, 0, stream>>>(x, bw, alpha, ws);
    lutgemm_bcq_reduce<<<NDIM / THREADS, THREADS, 0, stream>>>(ws, bias, out);

    (void)in_sizes; (void)n_in; (void)out_size; (void)ws_size;
}